// GlimpseSensor_33157147525409
// MI455X (gfx1250) — compile-verified
//
#include <hip/hip_runtime.h>
#include <hip/hip_bf16.h>
#include <stdint.h>

// Multi-scale glimpse sensor for MI455X (gfx1250), compile-only tuned.
//
// Roofline: ~12.6 MB read + 2.4 MB write vs ~75 MFLOP -> memory bound
// (~0.65 us at 23.3 TB/s). Strategy:
//  * One block per image. The three scale patches are concentric, so the
//    64x64x3 patch is DMA'd into LDS ONCE via global_load_async_to_lds_b32
//    (ASYNCcnt-tracked CDNA5 async path), zero-filling the pad halo.
//  * Each scale's antialiased-bilinear resize (jax.image.resize default,
//    triangle kernel of radius R=d/16) is a separable filter whose raw tap
//    weights are w_t = max(0, 1-|R-0.5-t|/R) -- independent of the output
//    index -- so templating on the scale makes all weights/pitches/trip
//    counts compile-time constants (no runtime div, fully unrolled taps).
//  * Edge taps are dropped and the partial weight sum renormalized, exactly
//    matching jax's compute_weight_mat.

#define IMG_H 224
#define IMG_W 224
#define IMG_C 3
#define G_WID 16
#define K_SC  3
#define PADW  32
#define MAXD  64
#define ROWL  (MAXD * IMG_C)       // 192 floats per big-patch row
#define P1    (ROWL + 4)           // patch pitch (pad to dodge 64-bank strides)
#define P2    49                   // hbuf pitch (odd -> bank friendly)

__device__ __forceinline__ unsigned lds_off32(const void* p) {
  // Flat LDS pointers carry the LDS byte offset in the low 32 bits
  // (ISA 10.2: LDS aperture).
  return (unsigned)(uintptr_t)p;
}

// Resample the centered DxD sub-patch of the 64x64 LDS patch down to 16x16
// for scale KS, separable triangle filter with 2R taps per axis.
template <int KS>
__device__ __forceinline__ void resample_scale(const float* __restrict__ patch,
                                               float* __restrict__ hbuf,
                                               float* __restrict__ op, int tid) {
  constexpr int   D    = G_WID << KS;        // 16 / 32 / 64
  constexpr int   R    = 1 << KS;            // downscale ratio
  constexpr int   T    = 2 * R;              // taps per axis
  constexpr int   OFF  = (MAXD - D) / 2;     // sub-patch origin: 24 / 16 / 0
  constexpr float INVR = 1.0f / (float)R;
  constexpr float CR   = (float)R - 0.5f;    // tap center: |CR - t| distances

  // ---- horizontal pass: D columns -> 16, per (row, out-col, channel) ----
  constexpr int NH = D * G_WID * IMG_C;
  for (int idx = tid; idx < NH; idx += 256) {
    const int r   = idx / (G_WID * IMG_C);
    const int rem = idx - r * (G_WID * IMG_C);   // j*3 + c
    const int j   = rem / IMG_C;
    const int c   = rem - j * IMG_C;
    const int i0  = R * j - (R >> 1);
    float acc = 0.0f, wsum = 0.0f;
#pragma unroll
    for (int t = 0; t < T; ++t) {
      const float w = 1.0f - fabsf(CR - (float)t) * INVR;  // folds to literal
      if (w > 0.0f) {                                      // folds (true for all t at every KS)
        const int i = i0 + t;
        if (i >= 0 && i < D) {                             // edge drop+renorm
          wsum += w;
          acc  += w * patch[(OFF + r) * P1 + (OFF + i) * IMG_C + c];
        }
      }
    }
    hbuf[r * P2 + rem] = acc / wsum;
  }
  __syncthreads();

  // ---- vertical pass: D rows -> 16, then coalesced global store ----
  for (int idx = tid; idx < G_WID * G_WID * IMG_C; idx += 256) {
    const int jy  = idx / (G_WID * IMG_C);
    const int rem = idx - jy * (G_WID * IMG_C);  // jx*3 + c
    const int i0  = R * jy - (R >> 1);
    float acc = 0.0f, wsum = 0.0f;
#pragma unroll
    for (int t = 0; t < T; ++t) {
      const float w = 1.0f - fabsf(CR - (float)t) * INVR;
      if (w > 0.0f) {
        const int i = i0 + t;
        if (i >= 0 && i < D) {
          wsum += w;
          acc  += w * hbuf[i * P2 + rem];
        }
      }
    }
    op[idx] = acc / wsum;
  }
  __syncthreads();   // hbuf reused by the next scale
}

__global__ __launch_bounds__(256) void glimpse_kernel(
    const float* __restrict__ img, const float* __restrict__ loc,
    float* __restrict__ out) {
  const int b   = blockIdx.x;
  const int tid = threadIdx.x;

  __shared__ float patch[MAXD * P1];   // 50176 B
  __shared__ float hbuf [MAXD * P2];   // 12544 B

  // ---- fixation center (reference floor/clip semantics) ----
  const float ly = loc[2 * b + 0];
  const float lx = loc[2 * b + 1];
  int cy = (int)floorf((ly + 1.0f) * 0.5f * (float)IMG_H);
  int cx = (int)floorf((lx + 1.0f) * 0.5f * (float)IMG_W);
  cy = min(max(cy, 0), IMG_H);
  cx = min(max(cx, 0), IMG_W);

  // 64x64 patch origin; dynamic_slice clamp included (provably inactive).
  int y0 = min(max(cy + PADW - MAXD / 2, 0), (IMG_H + 2 * PADW) - MAXD);
  int x0 = min(max(cx + PADW - MAXD / 2, 0), (IMG_W + 2 * PADW) - MAXD);
  const int ys = y0 - PADW;            // origin in unpadded image coords
  const int xs = x0 - PADW;

  const float* base = img + (size_t)b * (IMG_H * IMG_W * IMG_C);

  // ---- async DMA gather of the shared 64x64x3 patch; zero-fill halo ----
  for (int idx = tid; idx < MAXD * ROWL; idx += 256) {
    const int r   = idx / ROWL;                 // const-div
    const int rem = idx - r * ROWL;             // x*3 + c
    const int x   = rem / IMG_C;
    const int c   = rem - x * IMG_C;
    const int sy  = ys + r;
    const int sx  = xs + x;
    const unsigned lds = lds_off32(&patch[r * P1 + rem]);
    if (sy >= 0 && sy < IMG_H && sx >= 0 && sx < IMG_W) {
      const unsigned boff = (unsigned)(((sy * IMG_W + sx) * IMG_C + c) * 4);
      asm volatile("global_load_async_to_lds_b32 %0, %1, %2"
                   :: "v"(lds), "v"(boff), "s"(base)
                   : "memory");
    } else {
      patch[r * P1 + rem] = 0.0f;               // padded-image zeros
    }
  }
  asm volatile("s_wait_asynccnt 0x0" ::: "memory");  // our DMA landed
  __syncthreads();                                   // visible to all 8 waves

  // ---- three scales from the one staged patch ----
  float* ob = out + (size_t)b * (K_SC * G_WID * G_WID * IMG_C);
  resample_scale<0>(patch, hbuf, ob + 0 * (G_WID * G_WID * IMG_C), tid);
  resample_scale<1>(patch, hbuf, ob + 1 * (G_WID * G_WID * IMG_C), tid);
  resample_scale<2>(patch, hbuf, ob + 2 * (G_WID * G_WID * IMG_C), tid);
}

extern "C" void kernel_launch(void* const* d_in, const int* in_sizes, int n_in,
                              void* d_out, int out_size, void* d_ws, size_t ws_size,
                              hipStream_t stream) {
  const float* img = (const float*)d_in[0];
  const float* loc = (const float*)d_in[1];
  float* out = (float*)d_out;
  const int B = in_sizes[0] / (IMG_H * IMG_W * IMG_C);  // 256
  glimpse_kernel<<<dim3(B), 256, 0, stream>>>(img, loc, out);
}